// MAFA_78709570666778
// MI455X (gfx1250) — compile-verified
//
#include <hip/hip_runtime.h>
#include <hip/hip_bf16.h>

typedef __attribute__((ext_vector_type(2))) float v2f;
typedef __attribute__((ext_vector_type(4))) float v4f;
typedef __attribute__((ext_vector_type(8))) float v8f;

#define DIMC 192
#define HW   65536   // 256*256
#define WID  256

// ---------------------------------------------------------------------------
// K0: per-channel 8x8 circular-conv kernel from the rfft2 filter.
// y = irfft2(rfft2(x) * F)  ==  circular_conv(x, K),
// K[d,e] = (1/64) * sum_{u,v} F_full[u,v] * cos(2*pi*(u*d+v*e)/8)
// F_full[u,v] = F[u,v] (v<=4), F[(8-u)&7, 8-v] (v>=5).  Exact cos(pi/4*r) LUT.
// ---------------------------------------------------------------------------
__global__ void build_circ(const float* __restrict__ flt, float* __restrict__ circ) {
    const float ctab[8] = {1.f, 0.70710678118654752f, 0.f, -0.70710678118654752f,
                           -1.f, -0.70710678118654752f, 0.f, 0.70710678118654752f};
    int c = blockIdx.x;
    int t = threadIdx.x;          // 0..63
    int p = t >> 3, q = t & 7;
    const float* f = flt + c * 40;  // [8][5]
    float acc = 0.f;
    for (int u = 0; u < 8; ++u) {
        for (int v = 0; v < 8; ++v) {
            float F = (v <= 4) ? f[u * 5 + v] : f[((8 - u) & 7) * 5 + (8 - v)];
            acc += F * ctab[(u * p + v * q) & 7];
        }
    }
    circ[c * 64 + t] = acc * (1.f / 64.f);
}

// ---------------------------------------------------------------------------
// K1: channel-mix GEMM  Y1[b,o,n] = sum_c W[o,c] * X[b,c,n]  via
// V_WMMA_F32_16X16X4_F32.  Block = 256 threads = 8 waves (2 M-rows x 4 N-cols).
// Block tile: M=64, N=256.  Wave tile: 32 x 64 -> 8 v8f accumulators,
// 8 WMMAs per k-step fed by 2 A b64-loads + 8 B b32-loads.
// ---------------------------------------------------------------------------
__global__ __launch_bounds__(256) void proj_gemm(const float* __restrict__ X,
                                                 const float* __restrict__ Wt,
                                                 float* __restrict__ Y) {
    int bi  = blockIdx.x;
    int nt  = bi & 255;              // 256 N tiles of 256
    int tmp = bi >> 8;
    int mt  = tmp % 3;               // 3 M tiles of 64
    int b   = tmp / 3;

    int tid  = threadIdx.x;
    int wave = tid >> 5, lane = tid & 31;
    int l16  = lane & 15, hi = lane >> 4;

    int Mw = mt * 64 + (wave >> 2) * 32;        // wave covers M rows Mw..Mw+31
    int Nw = nt * 256 + (wave & 3) * 64;        // wave covers N cols Nw..Nw+63

    const float* Xb    = X + (size_t)b * DIMC * HW;
    const float* wrow0 = Wt + (size_t)(Mw + l16) * DIMC;       // M subtile 0
    const float* wrow1 = wrow0 + (size_t)16 * DIMC;            // M subtile 1

    v8f a00 = {}, a01 = {}, a02 = {}, a03 = {};
    v8f a10 = {}, a11 = {}, a12 = {}, a13 = {};

    for (int k0 = 0; k0 < DIMC; k0 += 4) {
        // A frags: 16x4 fp32, M = lane&15, K = k0 + 2*hi + {0,1}
        v2f fa0 = *(const v2f*)(wrow0 + k0 + 2 * hi);
        v2f fa1 = *(const v2f*)(wrow1 + k0 + 2 * hi);
        // B frags: 4x16, N = lane&15, K rows k0+2*hi and k0+2*hi+1
        const float* xr = Xb + (size_t)(k0 + 2 * hi) * HW + Nw + l16;
        v2f b0, b1, b2, b3;
        b0.x = xr[0];  b0.y = xr[HW + 0];
        b1.x = xr[16]; b1.y = xr[HW + 16];
        b2.x = xr[32]; b2.y = xr[HW + 32];
        b3.x = xr[48]; b3.y = xr[HW + 48];
        a00 = __builtin_amdgcn_wmma_f32_16x16x4_f32(false, fa0, false, b0, (short)0, a00, false, false);
        a01 = __builtin_amdgcn_wmma_f32_16x16x4_f32(false, fa0, false, b1, (short)0, a01, false, false);
        a02 = __builtin_amdgcn_wmma_f32_16x16x4_f32(false, fa0, false, b2, (short)0, a02, false, false);
        a03 = __builtin_amdgcn_wmma_f32_16x16x4_f32(false, fa0, false, b3, (short)0, a03, false, false);
        a10 = __builtin_amdgcn_wmma_f32_16x16x4_f32(false, fa1, false, b0, (short)0, a10, false, false);
        a11 = __builtin_amdgcn_wmma_f32_16x16x4_f32(false, fa1, false, b1, (short)0, a11, false, false);
        a12 = __builtin_amdgcn_wmma_f32_16x16x4_f32(false, fa1, false, b2, (short)0, a12, false, false);
        a13 = __builtin_amdgcn_wmma_f32_16x16x4_f32(false, fa1, false, b3, (short)0, a13, false, false);
    }

    float* Yb = Y + (size_t)b * DIMC * HW;
#pragma unroll
    for (int i = 0; i < 8; ++i) {
        int row0 = Mw + i + 8 * hi;          // C/D layout: vgpr i, M = i + 8*hi
        float* y0 = Yb + (size_t)row0 * HW + Nw + l16;
        y0[0]  = a00[i];  y0[16] = a01[i];  y0[32] = a02[i];  y0[48] = a03[i];
        float* y1 = y0 + (size_t)16 * HW;
        y1[0]  = a10[i];  y1[16] = a11[i];  y1[32] = a12[i];  y1[48] = a13[i];
    }
}

// ---------------------------------------------------------------------------
// K2: per-patch spectral filter as 64x64 circulant GEMM (in place on Y).
// One block = (b, c, 8-row strip) = 32 patches. LDS-staged patch matrix
// Xmat[k=64][patch=32] (pitch 33), A gathered from K_c circulant in LDS.
// Wide b128 staging loads and b128 epilogue stores.
// ---------------------------------------------------------------------------
__global__ __launch_bounds__(256) void spectral(float* __restrict__ Y,
                                                const float* __restrict__ circ) {
    __shared__ float Xmat[64 * 33];
    __shared__ float cs[64];

    int bi    = blockIdx.x;
    int strip = bi & 31;
    int c     = (bi >> 5) % DIMC;
    int b     = bi / (32 * DIMC);
    int tid   = threadIdx.x;

    float* base = Y + (size_t)(b * DIMC + c) * HW + (size_t)strip * 8 * WID;

    // stage strip (2048 floats) as 2 x b128 per thread, scatter into
    // Xmat[k=(s*8+t)][patch]
#pragma unroll
    for (int j = 0; j < 2; ++j) {
        int fi = tid + j * 256;          // float4 index 0..511
        int e  = fi * 4;                 // element index in strip
        v4f v  = *(const v4f*)(base + e);
        int s  = e >> 8;                 // strip row
        int col = e & 255;
        int patch = col >> 3, t = col & 7;   // t in {0,4}
        float* xm = &Xmat[(s * 8 + t) * 33 + patch];
        xm[0 * 33] = v.x; xm[1 * 33] = v.y; xm[2 * 33] = v.z; xm[3 * 33] = v.w;
    }
    if (tid < 64) cs[tid] = circ[c * 64 + tid];
    __syncthreads();

    int wave = tid >> 5, lane = tid & 31;
    int l16  = lane & 15, hi = lane >> 4;
    int mtb  = (wave >> 1) * 16;   // output-pixel tile base (M)
    int ntb  = (wave & 1) * 16;    // patch tile base (N)

    int m = mtb + l16;             // A layout: M = lane&15
    int p = m >> 3, q = m & 7;

    v8f acc = {};
    for (int k0 = 0; k0 < 64; k0 += 4) {
        int k  = k0 + 2 * hi;
        int s0 = k >> 3,        t0 = k & 7;
        int s1 = (k + 1) >> 3,  t1 = (k + 1) & 7;
        v2f a;
        a.x = cs[((p - s0) & 7) * 8 + ((q - t0) & 7)];
        a.y = cs[((p - s1) & 7) * 8 + ((q - t1) & 7)];
        v2f bf;
        bf.x = Xmat[k * 33 + ntb + l16];
        bf.y = Xmat[(k + 1) * 33 + ntb + l16];
        acc = __builtin_amdgcn_wmma_f32_16x16x4_f32(false, a, false, bf, (short)0, acc, false, false);
    }

    // epilogue: lane holds one full patch row (q = 0..7) -> 2 x b128 stores
    int po    = (mtb + 8 * hi) >> 3;
    int patch = ntb + l16;
    float* dst = base + po * WID + patch * 8;
    v4f lo, hv;
    lo.x = acc[0]; lo.y = acc[1]; lo.z = acc[2]; lo.w = acc[3];
    hv.x = acc[4]; hv.y = acc[5]; hv.z = acc[6]; hv.w = acc[7];
    *(v4f*)(dst + 0) = lo;
    *(v4f*)(dst + 4) = hv;
}

extern "C" void kernel_launch(void* const* d_in, const int* in_sizes, int n_in,
                              void* d_out, int out_size, void* d_ws, size_t ws_size,
                              hipStream_t stream) {
    const float* x   = (const float*)d_in[0];   // [8,192,256,256] f32
    const float* wpr = (const float*)d_in[1];   // [192,192] f32
    const float* flt = (const float*)d_in[2];   // [192,1,1,8,5] f32
    float* out  = (float*)d_out;                // [8,192,256,256] f32
    float* circ = (float*)d_ws;                 // 192*64 floats scratch

    build_circ<<<DIMC, 64, 0, stream>>>(flt, circ);
    proj_gemm<<<8 * 3 * 256, 256, 0, stream>>>(x, wpr, out);
    spectral<<<8 * DIMC * 32, 256, 0, stream>>>(out, circ);
}